// RNNLayer_74302934221332
// MI455X (gfx1250) — compile-verified
//
#include <hip/hip_runtime.h>
#include <math.h>

// MI455X / gfx1250, wave32. Matrix ops via V_WMMA_F32_16X16X4_F32 (fp32 native).

#define DEVI __device__ __forceinline__

typedef __attribute__((ext_vector_type(2))) float v2f;
typedef __attribute__((ext_vector_type(8))) float v8f;

constexpr int Bc = 128, Tc = 1024, Fc = 256, Uc = 256;
constexpr int HS = 260;  // padded LDS row stride (floats) -> conflict-free A reads

DEVI v8f wmma4(v2f a, v2f b, v8f c) {
  // 8 args: (neg_a, A, neg_b, B, c_mod, C, reuse_a, reuse_b); NEG must be 0 for f32
  return __builtin_amdgcn_wmma_f32_16x16x4_f32(false, a, false, b, (short)0, c,
                                               false, false);
}

// acc(16x16) += A(16xK, global, row stride lda) * W[:, n0:n0+16] (global, row stride ldb)
DEVI v8f gemm_global_a(const float* __restrict__ Arow0, int lda,
                       const float* __restrict__ Bn, int ldb, int K,
                       v8f acc, int l15, int half) {
#pragma unroll 4
  for (int kc = 0; kc < K; kc += 4) {
    v2f a = *(const v2f*)(Arow0 + (size_t)l15 * lda + kc + 2 * half);
    const float* pb = Bn + (size_t)(kc + 2 * half) * ldb + l15;
    v2f b;
    b.x = pb[0];
    b.y = pb[ldb];
    acc = wmma4(a, b, acc);
  }
  return acc;
}

// acc(16x16) += Hlds(16x256, LDS, padded stride HS) * W[:, n0:n0+16]
DEVI v8f gemm_lds_a(const float* hs, const float* __restrict__ Bn, int ldb,
                    v8f acc, int l15, int half) {
#pragma unroll 4
  for (int kc = 0; kc < Uc; kc += 4) {
    v2f a = *(const v2f*)(hs + l15 * HS + kc + 2 * half);
    const float* pb = Bn + (size_t)(kc + 2 * half) * ldb + l15;
    v2f b;
    b.x = pb[0];
    b.y = pb[ldb];
    acc = wmma4(a, b, acc);
  }
  return acc;
}

// ---- Kernel 1: xp[t][b][u] = (x @ enc_Wx + enc_b)  (fully parallel GEMM) ----
// xp is stored TRANSPOSED [T,B,U] so the recurrent kernel's per-step tile is a
// contiguous 16KB block. grid = (B*T)/16 blocks, 512 threads (16 waves).
__global__ __launch_bounds__(512) void xp_proj_kernel(
    const float* __restrict__ x, const float* __restrict__ Wx,
    const float* __restrict__ b, float* __restrict__ xp) {
  const int lane = threadIdx.x & 31;
  const int wave = threadIdx.x >> 5;
  const int l15 = lane & 15, half = lane >> 4;
  const int n0 = wave * 16;
  const size_t rbase = (size_t)blockIdx.x * 16;  // row in flattened [B*T, F]
  const int bidx = (int)(rbase / Tc);            // 16-row tile stays in one b
  const int tbase = (int)(rbase % Tc);

  const float bv = b[n0 + l15];
  v8f acc = {bv, bv, bv, bv, bv, bv, bv, bv};
  acc = gemm_global_a(x + rbase * Fc, Fc, Wx + n0, Uc, Fc, acc, l15, half);

#pragma unroll
  for (int r = 0; r < 8; ++r)
    xp[((size_t)(tbase + r + 8 * half) * Bc + bidx) * Uc + n0 + l15] = acc[r];
}

// ---------- Kernel 2: fused encoder+decoder recurrence, batch-parallel -------
// grid = B/16 = 8 blocks (independent recurrences per batch tile),
// 512 threads = 16 waves, wave w owns output N-tile w.
// Hidden states double-buffered in LDS -> only 2 barriers per timestep.
// Next step's xp fragment is preloaded into registers (software pipeline).
template <bool PRE>
__global__ __launch_bounds__(512) void rnn_fused_kernel(
    const float* __restrict__ xp, const float* __restrict__ x,
    const float* __restrict__ enc_Wx, const float* __restrict__ enc_b,
    const float* __restrict__ enc_Wh, const float* __restrict__ dec_Wx,
    const float* __restrict__ dec_Wh, const float* __restrict__ dec_b,
    float* __restrict__ out) {
  __shared__ float h_enc[2][16 * HS];
  __shared__ float h_dec[2][16 * HS];

  const int lane = threadIdx.x & 31;
  const int wave = threadIdx.x >> 5;
  const int l15 = lane & 15, half = lane >> 4;
  const int n0 = wave * 16;
  const int bbase = blockIdx.x * 16;  // batch rows [bbase, bbase+16)

  for (int i = threadIdx.x; i < 16 * HS; i += blockDim.x) {
    h_enc[0][i] = 0.0f;
    h_enc[1][i] = 0.0f;
    h_dec[0][i] = 0.0f;
    h_dec[1][i] = 0.0f;
  }
  __syncthreads();

  const float eb = enc_b[n0 + l15];
  const float db = dec_b[n0 + l15];
  v8f dacc = {};     // decoder state fragment (last step -> output)
  v8f acc_pre = {};  // pipelined xp fragment for the upcoming step

  if (PRE) {
    const float* p0 = xp + ((size_t)0 * Bc + bbase) * Uc + n0 + l15;
#pragma unroll
    for (int r = 0; r < 8; ++r) acc_pre[r] = p0[(r + 8 * half) * Uc];
  }

  for (int t = 0; t < Tc; ++t) {
    const int cur = t & 1, nxt = cur ^ 1;

    // ---- encoder step: h_enc' = tanh(xp_t + h_enc @ enc_Wh) ----
    v8f acc;
    if (PRE) {
      acc = acc_pre;
      if (t + 1 < Tc) {  // issue next step's xp loads early; consumed next iter
        const float* p = xp + ((size_t)(t + 1) * Bc + bbase) * Uc + n0 + l15;
#pragma unroll
        for (int r = 0; r < 8; ++r) acc_pre[r] = p[(r + 8 * half) * Uc];
      }
    } else {
      acc = (v8f){eb, eb, eb, eb, eb, eb, eb, eb};
      // on-the-fly input projection: A rows are batch rows -> stride T*F
      acc = gemm_global_a(x + (size_t)bbase * Tc * Fc + (size_t)t * Fc, Tc * Fc,
                          enc_Wx + n0, Uc, Fc, acc, l15, half);
    }
    acc = gemm_lds_a(h_enc[cur], enc_Wh + n0, Uc, acc, l15, half);
#pragma unroll
    for (int r = 0; r < 8; ++r) acc[r] = tanhf(acc[r]);
    // write to the OTHER buffer: no WAR hazard vs. waves still reading [cur]
#pragma unroll
    for (int r = 0; r < 8; ++r)
      h_enc[nxt][(r + 8 * half) * HS + n0 + l15] = acc[r];
    __syncthreads();  // h_enc[nxt] visible to all waves

    // ---- decoder step: h_dec' = tanh(h_enc' @ dec_Wx + dec_b + h_dec @ dec_Wh)
    v8f acc2 = {db, db, db, db, db, db, db, db};
    acc2 = gemm_lds_a(h_enc[nxt], dec_Wx + n0, Uc, acc2, l15, half);
    acc2 = gemm_lds_a(h_dec[cur], dec_Wh + n0, Uc, acc2, l15, half);
#pragma unroll
    for (int r = 0; r < 8; ++r) acc2[r] = tanhf(acc2[r]);
#pragma unroll
    for (int r = 0; r < 8; ++r)
      h_dec[nxt][(r + 8 * half) * HS + n0 + l15] = acc2[r];
    dacc = acc2;
    __syncthreads();  // h_dec[nxt] visible; protects next iteration's reads
  }

#pragma unroll
  for (int r = 0; r < 8; ++r)
    out[(size_t)(bbase + r + 8 * half) * Uc + n0 + l15] = dacc[r];
}

extern "C" void kernel_launch(void* const* d_in, const int* in_sizes, int n_in,
                              void* d_out, int out_size, void* d_ws,
                              size_t ws_size, hipStream_t stream) {
  const float* x = (const float*)d_in[0];
  const float* enc_Wx = (const float*)d_in[1];
  const float* enc_Wh = (const float*)d_in[2];
  const float* enc_b = (const float*)d_in[3];
  const float* dec_Wx = (const float*)d_in[4];
  const float* dec_Wh = (const float*)d_in[5];
  const float* dec_b = (const float*)d_in[6];
  float* out = (float*)d_out;

  const size_t xp_bytes = (size_t)Bc * Tc * Uc * sizeof(float);
  if (ws_size >= xp_bytes) {
    float* xp = (float*)d_ws;
    xp_proj_kernel<<<(Bc * Tc) / 16, 512, 0, stream>>>(x, enc_Wx, enc_b, xp);
    rnn_fused_kernel<true><<<Bc / 16, 512, 0, stream>>>(
        xp, x, enc_Wx, enc_b, enc_Wh, dec_Wx, dec_Wh, dec_b, out);
  } else {
    rnn_fused_kernel<false><<<Bc / 16, 512, 0, stream>>>(
        nullptr, x, enc_Wx, enc_b, enc_Wh, dec_Wx, dec_Wh, dec_b, out);
  }
}